// ChannelGate_25391846654427
// MI455X (gfx1250) — compile-verified
//
#include <hip/hip_runtime.h>
#include <math.h>

typedef float v2f __attribute__((ext_vector_type(2)));
typedef float v8f __attribute__((ext_vector_type(8)));

namespace {
constexpr int kB   = 32;
constexpr int kC   = 512;   // channels per half
constexpr int kCr  = 32;    // C // 16
constexpr int kHW  = 3136;  // 56*56
constexpr int kHW4 = 784;   // HW / 4
}

// ---------------------------------------------------------------------------
// Kernel 1: single fused pass over x (411 MB -> bandwidth bound).
// One block per (b, c) channel pair; computes mean(fr), mean(fi) and the
// first-argmax of score^2 = zr^2 + zi^2 (sqrt is monotonic -> same argmax),
// carrying the winning (fr, fi) values.
// ---------------------------------------------------------------------------
__global__ __launch_bounds__(256) void cg_reduce(
    const float* __restrict__ x,
    float* __restrict__ ar, float* __restrict__ ai,
    float* __restrict__ mr, float* __restrict__ mi)
{
  const int bc = blockIdx.x;           // b*512 + c
  const int b  = bc >> 9;
  const int c  = bc & 511;
  const float4* fr = (const float4*)(x + (size_t)(b * 1024 + c) * kHW);
  const float4* fi = (const float4*)(x + (size_t)(b * 1024 + 512 + c) * kHW);
  const int tid = threadIdx.x;

  float sr = 0.0f, si = 0.0f;
  float best = -__builtin_inff();
  float bfr = 0.0f, bfi = 0.0f;
  int   bidx = 0x7fffffff;

  for (int i = tid; i < kHW4; i += 256) {
    float4 r4 = fr[i];
    float4 i4 = fi[i];
    float rv[4] = {r4.x, r4.y, r4.z, r4.w};
    float iv[4] = {i4.x, i4.y, i4.z, i4.w};
#pragma unroll
    for (int j = 0; j < 4; ++j) {
      float r  = rv[j];
      float im = iv[j];
      sr += r;
      si += im;
      float d   = r * r + im * im;
      float inv = 1.0f / d;
      float zr  = r + r * inv;
      float zi  = im - im * inv;
      float s2  = zr * zr + zi * zi;
      int idx   = i * 4 + j;
      if (s2 > best || (s2 == best && idx < bidx)) {
        best = s2; bfr = r; bfi = im; bidx = idx;
      }
    }
  }

  __shared__ float s_sr[256], s_si[256], s_bs[256], s_br[256], s_bi[256];
  __shared__ int   s_ix[256];
  s_sr[tid] = sr;  s_si[tid] = si;
  s_bs[tid] = best; s_br[tid] = bfr; s_bi[tid] = bfi; s_ix[tid] = bidx;
  __syncthreads();

  for (int off = 128; off > 0; off >>= 1) {
    if (tid < off) {
      s_sr[tid] += s_sr[tid + off];
      s_si[tid] += s_si[tid + off];
      float os = s_bs[tid + off];
      int   oi = s_ix[tid + off];
      if (os > s_bs[tid] || (os == s_bs[tid] && oi < s_ix[tid])) {
        s_bs[tid] = os;
        s_br[tid] = s_br[tid + off];
        s_bi[tid] = s_bi[tid + off];
        s_ix[tid] = oi;
      }
    }
    __syncthreads();
  }

  if (tid == 0) {
    ar[bc] = s_sr[0] * (1.0f / kHW);
    ai[bc] = s_si[0] * (1.0f / kHW);
    mr[bc] = s_br[0];
    mi[bc] = s_bi[0];
  }
}

// Load a 2-element (per-lane) f32 WMMA fragment pair from a row-major matrix.
__device__ __forceinline__ v2f frag2(const float* __restrict__ p, int ld, int row, int k) {
  v2f f;
  f.x = p[row * ld + k];
  f.y = p[row * ld + k + 1];
  return f;
}

// ---------------------------------------------------------------------------
// Kernel 2: att = mlp(ar,ai) + mlp(mr,mi) via V_WMMA_F32_16X16X4_F32.
// One block, 8 waves. Layer 1 (M32,N32,K512): one 16x16 tile-job per wave.
// Layer 2 (M32,N512,K32): 128 tile-jobs, 16 per wave. Hidden activations +
// cardioid staged in LDS. Pass 0 writes d_out, pass 1 accumulates (same wave
// owns the same tiles in both passes -> no race).
// ---------------------------------------------------------------------------
__global__ __launch_bounds__(256) void cg_mlp(
    const float* __restrict__ ar, const float* __restrict__ ai,
    const float* __restrict__ mr, const float* __restrict__ mi,
    const float* __restrict__ w1r, const float* __restrict__ b1r,
    const float* __restrict__ w1i, const float* __restrict__ b1i,
    const float* __restrict__ w2r, const float* __restrict__ b2r,
    const float* __restrict__ w2i, const float* __restrict__ b2i,
    float* __restrict__ out)
{
  __shared__ float sh_hr[32 * 32];
  __shared__ float sh_hi[32 * 32];

  const int tid  = threadIdx.x;
  const int wave = tid >> 5;
  const int lane = tid & 31;
  const int ln   = lane & 15;          // tile column for B/C/D frags; tile row for A
  const int hi8  = (lane >> 4) << 3;   // C/D row offset: 0 or 8
  const int k2   = (lane >> 4) << 1;   // A/B K offset: 0 or 2

  for (int pass = 0; pass < 2; ++pass) {
    const float* reIn = pass ? mr : ar;
    const float* imIn = pass ? mi : ai;

    // ---- layer 1: job == wave (0..7): {hr|hi} x mt x nt -------------------
    {
      const int outIdx = wave >> 2;            // 0 = hr, 1 = hi
      const int mt = (wave >> 1) & 1;
      const int nt = wave & 1;
      const int n  = nt * 16 + ln;
      const float bias = outIdx ? (b1r[n] + b1i[n]) : (b1r[n] - b1i[n]);
      const float* A1 = outIdx ? imIn : reIn;  // product with w1r
      const float* A2 = outIdx ? reIn : imIn;  // product with w1i (signed)
      const float sg  = outIdx ? 1.0f : -1.0f;
      v8f acc;
#pragma unroll
      for (int j = 0; j < 8; ++j) acc[j] = bias;
      const int arow = mt * 16 + ln;
      for (int k0 = 0; k0 < kC; k0 += 4) {
        v2f a1  = frag2(A1,  kC, arow, k0 + k2);
        v2f a2  = frag2(A2,  kC, arow, k0 + k2);
        a2.x *= sg; a2.y *= sg;
        v2f br_ = frag2(w1r, kC, n, k0 + k2);   // B[k][n] = w1r[n][k]
        v2f bi_ = frag2(w1i, kC, n, k0 + k2);
        acc = __builtin_amdgcn_wmma_f32_16x16x4_f32(false, a1, false, br_,
                                                    (short)0, acc, false, false);
        acc = __builtin_amdgcn_wmma_f32_16x16x4_f32(false, a2, false, bi_,
                                                    (short)0, acc, false, false);
      }
      float* dst = outIdx ? sh_hi : sh_hr;
      const int rb = mt * 16 + hi8;
#pragma unroll
      for (int j = 0; j < 8; ++j) dst[(rb + j) * 32 + n] = acc[j];
    }
    __syncthreads();

    // ---- cardioid: s = 0.5*(1 + cos(atan2(hi,hr))) = 0.5*(1 + hr/|h|) -----
    for (int i = tid; i < 1024; i += 256) {
      float r  = sh_hr[i];
      float im = sh_hi[i];
      float mag = sqrtf(r * r + im * im);
      float s = (mag > 0.0f) ? (0.5f * (1.0f + r / mag)) : 1.0f;
      sh_hr[i] = r * s;
      sh_hi[i] = im * s;
    }
    __syncthreads();

    // ---- layer 2: 128 jobs: {re|im} x mt x nt(0..31) ----------------------
    for (int job = wave; job < 128; job += 8) {
      const int outIdx = job >> 6;             // 0 = out_re, 1 = out_im
      const int mt = (job >> 5) & 1;
      const int nt = job & 31;
      const int n  = nt * 16 + ln;
      const float bias = outIdx ? (b2r[n] + b2i[n]) : (b2r[n] - b2i[n]);
      const float* A1 = outIdx ? sh_hi : sh_hr;  // product with w2r
      const float* A2 = outIdx ? sh_hr : sh_hi;  // product with w2i (signed)
      const float sg  = outIdx ? 1.0f : -1.0f;
      v8f acc;
#pragma unroll
      for (int j = 0; j < 8; ++j) acc[j] = bias;
      const int arow = mt * 16 + ln;
#pragma unroll
      for (int k0 = 0; k0 < kCr; k0 += 4) {
        v2f a1  = frag2(A1,  kCr, arow, k0 + k2);
        v2f a2  = frag2(A2,  kCr, arow, k0 + k2);
        a2.x *= sg; a2.y *= sg;
        v2f br_ = frag2(w2r, kCr, n, k0 + k2);  // B[k][n] = w2r[n][k]
        v2f bi_ = frag2(w2i, kCr, n, k0 + k2);
        acc = __builtin_amdgcn_wmma_f32_16x16x4_f32(false, a1, false, br_,
                                                    (short)0, acc, false, false);
        acc = __builtin_amdgcn_wmma_f32_16x16x4_f32(false, a2, false, bi_,
                                                    (short)0, acc, false, false);
      }
      const int colg = outIdx * 512 + n;       // concat([re, im], axis=1)
      const int rb   = mt * 16 + hi8;
      if (pass == 0) {
#pragma unroll
        for (int j = 0; j < 8; ++j) out[(rb + j) * 1024 + colg] = acc[j];
      } else {
#pragma unroll
        for (int j = 0; j < 8; ++j) {
          int o = (rb + j) * 1024 + colg;
          out[o] = out[o] + acc[j];
        }
      }
    }
    __syncthreads();
  }
}

extern "C" void kernel_launch(void* const* d_in, const int* in_sizes, int n_in,
                              void* d_out, int out_size, void* d_ws, size_t ws_size,
                              hipStream_t stream) {
  const float* x   = (const float*)d_in[0];
  const float* w1r = (const float*)d_in[1];
  const float* b1r = (const float*)d_in[2];
  const float* w1i = (const float*)d_in[3];
  const float* b1i = (const float*)d_in[4];
  const float* w2r = (const float*)d_in[5];
  const float* b2r = (const float*)d_in[6];
  const float* w2i = (const float*)d_in[7];
  const float* b2i = (const float*)d_in[8];

  float* ws = (float*)d_ws;                 // 4 * 16384 floats = 256 KB
  float* ar = ws;
  float* ai = ar + kB * kC;
  float* mr = ai + kB * kC;
  float* mi = mr + kB * kC;

  cg_reduce<<<dim3(kB * kC), dim3(256), 0, stream>>>(x, ar, ai, mr, mi);
  cg_mlp<<<dim3(1), dim3(256), 0, stream>>>(ar, ai, mr, mi,
                                            w1r, b1r, w1i, b1i,
                                            w2r, b2r, w2i, b2i,
                                            (float*)d_out);
}